// RelGraphConvLayer_32521492365634
// MI455X (gfx1250) — compile-verified
//
#include <hip/hip_runtime.h>
#include <stdint.h>

// ---------------------------------------------------------------------------
// RGCN layer for MI455X (gfx1250, wave32).
//   h = relu( sum_r (scatter_mean_r(x)) @ W_r  +  x @ W_loop + bias )
// Bandwidth-bound (~0.5-0.9 GB traffic @ 23.3 TB/s); GEMM is only 16.4 GFLOP,
// so exact fp32 WMMA (V_WMMA_F32_16X16X4_F32) is used. A-tiles are staged
// into LDS with CDNA5 async DMA (GLOBAL_LOAD_ASYNC_TO_LDS_B128 + ASYNCcnt).
// ---------------------------------------------------------------------------

typedef float v2f __attribute__((ext_vector_type(2)));
typedef float v8f __attribute__((ext_vector_type(8)));

#define NNODES 100000
#define DIM    128
#define NREL   4
#define KTOT   ((NREL + 1) * DIM)   // 640 concatenated K
#define LDA    (KTOT + 4)           // padded LDS row stride (4-float skew)

// One CDNA5 async DMA: 32 lanes x 16 B = 512 B global -> LDS, ASYNCcnt-tracked.
__device__ __forceinline__ void async_copy16(uint32_t lds_byte_addr,
                                             const void* gaddr) {
  asm volatile("global_load_async_to_lds_b128 %0, %1, off"
               :: "v"(lds_byte_addr), "v"(gaddr)
               : "memory");
}

// ---------------------------------------------------------------------------
// Kernel 1: zero agg + deg workspace (required every call).
// ---------------------------------------------------------------------------
__global__ __launch_bounds__(256) void zero_ws_kernel(float4* __restrict__ ws,
                                                      long n4) {
  long i = (long)blockIdx.x * 256 + threadIdx.x;
  if (i < n4) ws[i] = make_float4(0.f, 0.f, 0.f, 0.f);
}

// ---------------------------------------------------------------------------
// Kernel 2: edge scatter. Grid = (E/8, R). One wave32 per edge; lane L owns
// cols 4L..4L+3 (float4 load, 4x global_atomic_add_f32).
// ---------------------------------------------------------------------------
__global__ __launch_bounds__(256) void scatter_edges_kernel(
    const float* __restrict__ x, const int* __restrict__ src,
    const int* __restrict__ dst, float* __restrict__ agg,
    float* __restrict__ deg, int nedge) {
  const int wid  = threadIdx.x >> 5;
  const int lane = threadIdx.x & 31;
  const int e = blockIdx.x * 8 + wid;
  const int r = blockIdx.y;
  if (e >= nedge) return;

  const long slot = (long)r * nedge + e;
  const int s = src[slot];
  const int d = dst[slot];

  const float4 v = *(const float4*)(x + (long)s * DIM + lane * 4);
  float* base = agg + ((long)r * NNODES + d) * DIM + lane * 4;
  unsafeAtomicAdd(base + 0, v.x);
  unsafeAtomicAdd(base + 1, v.y);
  unsafeAtomicAdd(base + 2, v.z);
  unsafeAtomicAdd(base + 3, v.w);
  if (lane == 0) unsafeAtomicAdd(deg + (long)r * NNODES + d, 1.0f);
}

// ---------------------------------------------------------------------------
// Kernel 3: fused normalize + 5-way GEMM (K=640) + bias + ReLU.
// Block = 256 threads (8 waves) owns a 16x128 output tile; wave w owns cols
// [16w,16w+16). Raw A rows (agg & x) are DMA'd into LDS asynchronously; the
// 1/deg normalization is folded into the A-fragment load with a per-relation
// compile-time-indexed scale register (no cndmask selects). B (weights,
// 320 KB total) reads stay in WGP$/L2.
// ---------------------------------------------------------------------------
__global__ __launch_bounds__(256) void rgcn_gemm_kernel(
    const float* __restrict__ x, const float* __restrict__ agg,
    const float* __restrict__ deg, const float* __restrict__ relW,
    const float* __restrict__ loopW, const float* __restrict__ bias,
    float* __restrict__ out) {
  __shared__ float Atile[16 * LDA];
  __shared__ float invdeg[NREL][16];

  const int row0 = blockIdx.x * 16;             // N % 16 == 0 (6250 tiles)
  const int t = threadIdx.x;
  const int wid = t >> 5, lane = t & 31;
  const int lo = lane & 15;                     // row (A) / col (B,D) in tile
  const int hi = lane >> 4;                     // K-pair selector

  // --- Stage A = [agg_0|agg_1|agg_2|agg_3|x] (16 x 640 f32) via async DMA.
  // 80 segments of 512 B (row m, source r); wave w issues segments w+8j.
  const uint32_t lds_base = (uint32_t)(uintptr_t)Atile;
#pragma unroll
  for (int j = 0; j < 10; ++j) {
    const int seg = wid + 8 * j;
    const int m = seg / 5;                      // tile row
    const int r = seg % 5;                      // source: 0..3 agg, 4 = x
    const int n = row0 + m;
    const float* gsrc = (r < NREL) ? (agg + ((long)r * NNODES + n) * DIM)
                                   : (x + (long)n * DIM);
    const uint32_t ldsa =
        lds_base + (uint32_t)(m * LDA + r * DIM + lane * 4) * 4u;
    async_copy16(ldsa, gsrc + lane * 4);
  }

  // 1/max(deg,1) per (relation, row) while the DMAs are in flight.
  if (t < NREL * 16) {
    const int r = t >> 4, m = t & 15;
    const float dg = deg[(long)r * NNODES + row0 + m];
    invdeg[r][m] = 1.f / fmaxf(dg, 1.f);
  }

  asm volatile("s_wait_asynccnt 0" ::: "memory");
  __syncthreads();

  float invs[NREL];
#pragma unroll
  for (int r = 0; r < NREL; ++r) invs[r] = invdeg[r][lo];

  const int nb = wid * 16;                      // wave's output column base
  v8f c = {};                                   // 16x16 f32 accumulator
  const float* ApBase = Atile + lo * LDA + 2 * hi;
  const float* pRel  = relW  + (2 * hi) * DIM + nb + lo;   // B rows 2hi,2hi+1
  const float* pLoop = loopW + (2 * hi) * DIM + nb + lo;

  // Relation K-ranges: outer loop fully unrolled so the normalization scale
  // is a fixed register per 128-K span (no runtime-indexed selects).
#pragma unroll
  for (int r = 0; r < NREL; ++r) {
    const float s = invs[r];
#pragma unroll 8
    for (int k0 = r * DIM; k0 < (r + 1) * DIM; k0 += 4) {
      v2f a = *(const v2f*)(ApBase + k0);       // ds_load_b64 (pairs merge)
      a *= s;                                   // fused mean-normalization
      v2f b;
      b.x = pRel[(long)k0 * DIM];
      b.y = pRel[(long)k0 * DIM + DIM];
      c = __builtin_amdgcn_wmma_f32_16x16x4_f32(
          false, a, false, b, (short)0, c, false, false);
    }
  }
  // Self-loop K-range [512,640): x @ loop_weight, unscaled.
#pragma unroll 8
  for (int k0 = 0; k0 < DIM; k0 += 4) {
    const v2f a = *(const v2f*)(ApBase + NREL * DIM + k0);
    v2f b;
    b.x = pLoop[(long)k0 * DIM];
    b.y = pLoop[(long)k0 * DIM + DIM];
    c = __builtin_amdgcn_wmma_f32_16x16x4_f32(
        false, a, false, b, (short)0, c, false, false);
  }

  // Epilogue: bias + ReLU. C/D layout: VGPR v -> row v + 8*hi, col lo.
  const float bcol = bias[nb + lo];
#pragma unroll
  for (int v = 0; v < 8; ++v) {
    const int n = row0 + v + 8 * hi;
    const float h = c[v] + bcol;
    out[(long)n * DIM + nb + lo] = fmaxf(h, 0.f);
  }
}

// ---------------------------------------------------------------------------
// Launch. Inputs: 0=x[N,128] f32, 1=src_idx[R,E] i32, 2=dst_idx[R,E] i32,
// 3=rel_weight[R,128,128] f32, 4=loop_weight[128,128] f32, 5=h_bias[128] f32.
// Workspace: agg (R*N*D f32) then deg (R*N f32) ~= 206.4 MB.
// ---------------------------------------------------------------------------
extern "C" void kernel_launch(void* const* d_in, const int* in_sizes, int n_in,
                              void* d_out, int out_size, void* d_ws, size_t ws_size,
                              hipStream_t stream) {
  const float* x     = (const float*)d_in[0];
  const int*   src   = (const int*)d_in[1];
  const int*   dst   = (const int*)d_in[2];
  const float* relW  = (const float*)d_in[3];
  const float* loopW = (const float*)d_in[4];
  const float* bias  = (const float*)d_in[5];
  float* out = (float*)d_out;

  const int nedge = in_sizes[1] / NREL;         // E = 150000

  float* agg = (float*)d_ws;
  float* deg = agg + (long)NREL * NNODES * DIM;

  // 1) zero agg + deg
  const long n4 = ((long)NREL * NNODES * DIM + (long)NREL * NNODES) / 4;
  const int zblocks = (int)((n4 + 255) / 256);
  zero_ws_kernel<<<zblocks, 256, 0, stream>>>((float4*)d_ws, n4);

  // 2) scatter-sum edges: grid (E/8, R), 8 edges per 256-thread block
  dim3 sgrid((nedge + 7) / 8, NREL);
  scatter_edges_kernel<<<sgrid, 256, 0, stream>>>(x, src, dst, agg, deg, nedge);

  // 3) fused normalize + GEMM + bias + ReLU (6250 row tiles)
  const int gblocks = (NNODES + 15) / 16;
  rgcn_gemm_kernel<<<gblocks, 256, 0, stream>>>(x, agg, deg, relW, loopW, bias, out);
}